// RegularSimilar_2886218023070
// MI455X (gfx1250) — compile-verified
//
#include <hip/hip_runtime.h>
#include <hip/hip_bf16.h>

// Problem constants (from the reference)
#define B_SZ   4096
#define D_SZ   256
#define S_SZ   256
#define K2D    512   // 2*D
#define KTOT   513   // 2*D + 1 (privacy column)

typedef __attribute__((ext_vector_type(2))) float v2f;
typedef __attribute__((ext_vector_type(8))) float v8f;

// ---------------------------------------------------------------------------
// Kernel 1: user_item_feature = [union_feature | privacy] @ W^T + bias
//   (4096 x 513) @ (513 x 256) fp32 GEMM on V_WMMA_F32_16X16X4_F32.
// One wave computes one 16x16 output tile. K loop covers k=0..511 in steps
// of 4; the k=512 privacy column and the bias are folded into the epilogue so
// the hot loop has no bounds checks (EXEC stays all-ones for WMMA).
// ---------------------------------------------------------------------------
__global__ __launch_bounds__(128) void uif_gemm_kernel(
    const float* __restrict__ unionf,   // (B, 512)
    const float* __restrict__ priv,     // (B,)
    const float* __restrict__ W,        // (256, 513) row-major (d, k)
    const float* __restrict__ bias,     // (256,)
    float* __restrict__ uif)            // (B, 256) out
{
    const int lane  = threadIdx.x & 31;
    const int wave  = threadIdx.x >> 5;
    const int bTile = blockIdx.x;              // 0..255
    const int dTile = blockIdx.y * 4 + wave;   // 0..15
    const int bm = bTile * 16;
    const int dn = dTile * 16;

    // ISA layout (16x4 fp32 A / 4x16 fp32 B):
    //   lane L: m/n = L%16; lanes 0-15 hold K={0,1}, lanes 16-31 hold K={2,3}
    const int mn   = lane & 15;
    const int hi   = lane >> 4;
    const int koff = hi * 2;

    const float* arow = unionf + (size_t)(bm + mn) * K2D;  // A row (m = mn)
    const float* bcol = W      + (size_t)(dn + mn) * KTOT; // B col (n = mn) = W row

    v8f c = {};
    for (int kk = 0; kk < K2D; kk += 4) {
        const int k0 = kk + koff;
        v2f a, bv;
        a.x  = arow[k0];
        a.y  = arow[k0 + 1];
        bv.x = bcol[k0];
        bv.y = bcol[k0 + 1];
#if __has_builtin(__builtin_amdgcn_wmma_f32_16x16x4_f32)
        // 8 args: (neg_a, A, neg_b, B, c_mod, C, reuse_a, reuse_b)
        c = __builtin_amdgcn_wmma_f32_16x16x4_f32(false, a, false, bv,
                                                  (short)0, c, false, false);
#else
        // Fallback so the TU still compiles if the builtin is absent
        // (not numerically the tile product; probe says builtin exists).
        c[0] = fmaf(a.x, bv.x, fmaf(a.y, bv.y, c[0]));
#endif
    }

    // Epilogue: k==512 (privacy column) + bias, then store the 16x16 tile.
    // C/D layout: VGPR r -> row m = r + hi*8, col n = mn.
    const float wlast = bcol[512];
    const float bn    = bias[dn + mn];
#pragma unroll
    for (int r = 0; r < 8; ++r) {
        const int m = r + hi * 8;
        const float val = c[r] + priv[bm + m] * wlast + bn;
        uif[(size_t)(bm + m) * D_SZ + (dn + mn)] = val;
    }
}

// ---------------------------------------------------------------------------
// Kernel 2: per-b scores + softmax.
// One 256-thread block (8 waves) per b. Each wave computes the score of one
// sampled row per iteration: lanes read a contiguous 1KB embedding row
// (coalesced, L2-resident since all_items fits in the 192MB L2), FMA against
// the cached uif row, then shfl_xor tree-reduce across the wave32.
// Then a block softmax over the 256 scores; probs go to scratch and the
// truncated weighted index sum to d_out[0..B).
// ---------------------------------------------------------------------------
__global__ __launch_bounds__(256) void score_softmax_kernel(
    const int*   __restrict__ need_replace,      // (B, 2)
    const int*   __restrict__ user_sample_items, // (N_USERS, 256)
    const float* __restrict__ all_items,         // (N_ITEMS, 256)
    const float* __restrict__ uif,               // (B, 256)
    float*       __restrict__ probs,             // (B, 256) scratch
    float*       __restrict__ out_items)         // d_out[0..B)
{
    __shared__ int   sid[S_SZ];
    __shared__ float sc[S_SZ];
    __shared__ float red[S_SZ];

    const int b    = blockIdx.x;
    const int t    = threadIdx.x;
    const int lane = t & 31;
    const int wave = t >> 5;

    const int uid = need_replace[2 * b];
    sid[t] = user_sample_items[(size_t)uid * S_SZ + t];

    // uif row cached in registers: lane owns d = j*32 + lane
    float u[8];
    const float* urow = uif + (size_t)b * D_SZ;
#pragma unroll
    for (int j = 0; j < 8; ++j) u[j] = urow[j * 32 + lane];
    __syncthreads();

    for (int it = 0; it < S_SZ / 8; ++it) {
        const int s = it * 8 + wave;
        const float* erow = all_items + (size_t)sid[s] * D_SZ;
        float acc = 0.f;
#pragma unroll
        for (int j = 0; j < 8; ++j) acc = fmaf(u[j], erow[j * 32 + lane], acc);
#pragma unroll
        for (int off = 16; off > 0; off >>= 1)
            acc += __shfl_xor(acc, off, 32);
        if (lane == 0) sc[s] = acc;
    }
    __syncthreads();

    // Block softmax over 256 scores
    const float v = sc[t];
    red[t] = v; __syncthreads();
    for (int off = 128; off > 0; off >>= 1) {
        if (t < off) red[t] = fmaxf(red[t], red[t + off]);
        __syncthreads();
    }
    const float mx = red[0];
    __syncthreads();

    const float e = __expf(v - mx);
    red[t] = e; __syncthreads();
    for (int off = 128; off > 0; off >>= 1) {
        if (t < off) red[t] = red[t] + red[t + off];
        __syncthreads();
    }
    const float denom = red[0];
    __syncthreads();

    const float p = e / denom;
    probs[(size_t)b * S_SZ + t] = p;

    // replaceable_items[b] = int32(sum(p * float(id)))  (truncation toward 0)
    red[t] = p * (float)sid[t]; __syncthreads();
    for (int off = 128; off > 0; off >>= 1) {
        if (t < off) red[t] = red[t] + red[t + off];
        __syncthreads();
    }
    if (t == 0) out_items[b] = (float)(int)red[0];
}

// ---------------------------------------------------------------------------
// Kernel 3: replaceable_items_feature[b, d] = sum_s p[s] * emb[sid[s], d]
// One 256-thread block per b; thread d accumulates over all 256 rows, each
// row read is a fully-coalesced 1KB burst (hot in L2 after kernel 2).
// Also zeros the two scalar outputs.
// ---------------------------------------------------------------------------
__global__ __launch_bounds__(256) void weighted_feat_kernel(
    const int*   __restrict__ need_replace,
    const int*   __restrict__ user_sample_items,
    const float* __restrict__ all_items,
    const float* __restrict__ probs,
    float*       __restrict__ out_feat,     // d_out + B, (B, 256)
    float*       __restrict__ out_scalars)  // d_out + B + B*D, 2 floats
{
    __shared__ int   sid[S_SZ];
    __shared__ float p[S_SZ];

    const int b = blockIdx.x;
    const int t = threadIdx.x;

    const int uid = need_replace[2 * b];
    sid[t] = user_sample_items[(size_t)uid * S_SZ + t];
    p[t]   = probs[(size_t)b * S_SZ + t];
    __syncthreads();

    float acc = 0.f;
#pragma unroll 4
    for (int s = 0; s < S_SZ; ++s)
        acc = fmaf(p[s], all_items[(size_t)sid[s] * D_SZ + t], acc);

    out_feat[(size_t)b * D_SZ + t] = acc;

    if (b == 0 && t < 2) out_scalars[t] = 0.f;  // similarity_loss, similarity
}

// ---------------------------------------------------------------------------
extern "C" void kernel_launch(void* const* d_in, const int* in_sizes, int n_in,
                              void* d_out, int out_size, void* d_ws, size_t ws_size,
                              hipStream_t stream) {
    const int*   need_replace = (const int*)  d_in[0]; // (B,2)
    const float* unionf       = (const float*)d_in[1]; // (B,512)
    const float* all_items    = (const float*)d_in[2]; // (100000,256)
    const float* priv         = (const float*)d_in[3]; // (B,)
    const int*   usi          = (const int*)  d_in[4]; // (20000,256)
    const float* W            = (const float*)d_in[5]; // (256,513)
    const float* bias         = (const float*)d_in[6]; // (256,)

    float* out = (float*)d_out;

    // Scratch layout: [uif : B*D floats][probs : B*S floats] = 8 MB total
    float* uif   = (float*)d_ws;
    float* probs = uif + (size_t)B_SZ * D_SZ;

    // K1: WMMA fp32 GEMM, 4096 tiles, 1 wave each (128-thread blocks = 4 waves)
    dim3 g1(B_SZ / 16, 4), b1(128);
    hipLaunchKernelGGL(uif_gemm_kernel, g1, b1, 0, stream,
                       unionf, priv, W, bias, uif);

    // K2: scores + softmax, one block per b
    hipLaunchKernelGGL(score_softmax_kernel, dim3(B_SZ), dim3(256), 0, stream,
                       need_replace, usi, all_items, uif, probs, out);

    // K3: weighted feature, one block per b
    hipLaunchKernelGGL(weighted_feat_kernel, dim3(B_SZ), dim3(256), 0, stream,
                       need_replace, usi, all_items, probs,
                       out + B_SZ, out + B_SZ + (size_t)B_SZ * D_SZ);
}